// RWKV_RNN_34282428956837
// MI455X (gfx1250) — compile-verified
//
#include <hip/hip_runtime.h>
#include <cstddef>
#include <cstdint>

// RWKV-v4 single-token inference, MI455X (gfx1250).
// Bandwidth-bound: ~1.46 GB fp32 weights/token @ 23.3 TB/s => ~63us floor, so
// stay fp32 and stream weights once. Matvecs use V_WMMA_F32_16X16X4_F32 (one
// wave = 16 output rows, K-reduction in the matrix pipe). The x vector is
// staged into LDS with GLOBAL_LOAD_ASYNC_TO_LDS_B128 (ASYNCcnt). All serial
// per-layer vector work (LN, token-shift mixes, WKV recurrence) is fused into
// the LAST arriving block of the preceding matvec (fence + atomic counter;
// non-last blocks exit, so no residency assumption) => 4 launches per layer.

#define Lh 24
#define Eh 1024
#define Hh 4096
#define Vh 50277

typedef float v2f __attribute__((ext_vector_type(2)));
typedef float v8f __attribute__((ext_vector_type(8)));

// ---------- async memory->LDS staging of the x vector (16B per transfer) ----------
__device__ __forceinline__ void stage_x_async(const float* __restrict__ x,
                                              float* sxv, int K)
{
    const int t = threadIdx.x;
    unsigned lds = (unsigned)(uintptr_t)sxv + (unsigned)t * 16u;  // LDS byte addr
    for (int off = t * 16; off < K * 4; off += 256 * 16) {
        asm volatile("global_load_async_to_lds_b128 %0, %1, %2"
                     :: "v"(lds), "v"(off), "s"(x) : "memory");
        lds += 256u * 16u;
    }
    asm volatile("s_wait_asynccnt 0x0" ::: "memory");
    __syncthreads();
}

// ---------- "last block" grid fan-in (no spin, non-last blocks exit) ----------
__device__ __forceinline__ bool grid_sink(unsigned* cnt, unsigned nblocks)
{
    __shared__ unsigned lastf;
    __threadfence();            // make this block's global writes visible
    __syncthreads();
    if (threadIdx.x == 0) {
        unsigned prev = atomicAdd(cnt, 1u);
        unsigned l = (prev == nblocks - 1u) ? 1u : 0u;
        if (l) *cnt = 0u;       // reset for the next launch that reuses it
        lastf = l;
    }
    __syncthreads();
    if (!lastf) return false;
    __threadfence();            // acquire side: observe all blocks' writes
    return true;
}

// ---------- LayerNorm over 1024 channels with 256 threads (4 per thread) ----------
__device__ __forceinline__ void ln1024_t256(const float* __restrict__ x,
                                            const float* __restrict__ w,
                                            const float* __restrict__ b,
                                            float xn[4])
{
    __shared__ float r1[256];
    __shared__ float r2[256];
    const int t = threadIdx.x;
    float xi[4], s = 0.f, s2 = 0.f;
    #pragma unroll
    for (int j = 0; j < 4; ++j) {
        xi[j] = x[t + 256 * j];
        s += xi[j]; s2 += xi[j] * xi[j];
    }
    r1[t] = s; r2[t] = s2;
    __syncthreads();
    for (int st = 128; st > 0; st >>= 1) {
        if (t < st) { r1[t] += r1[t + st]; r2[t] += r2[t + st]; }
        __syncthreads();
    }
    const float mean = r1[0] * (1.0f / 1024.0f);
    const float var  = r2[0] * (1.0f / 1024.0f) - mean * mean;
    const float rstd = rsqrtf(var + 1e-5f);
    #pragma unroll
    for (int j = 0; j < 4; ++j)
        xn[j] = (xi[j] - mean) * rstd * w[t + 256 * j] + b[t + 256 * j];
}

// ---------- LayerNorm over 1024 channels with 1024 threads ----------
__device__ __forceinline__ float block_ln_1024(float xi, const float* __restrict__ w,
                                               const float* __restrict__ b, int i)
{
    __shared__ float s_sum[1024];
    __shared__ float s_sq[1024];
    __syncthreads();            // WAR guard for back-to-back calls
    s_sum[i] = xi; s_sq[i] = xi * xi;
    __syncthreads();
    for (int s = 512; s > 0; s >>= 1) {
        if (i < s) { s_sum[i] += s_sum[i + s]; s_sq[i] += s_sq[i + s]; }
        __syncthreads();
    }
    const float mean = s_sum[0] * (1.0f / 1024.0f);
    const float var  = s_sq[0] * (1.0f / 1024.0f) - mean * mean;
    return (xi - mean) * rsqrtf(var + 1e-5f) * w[i] + b[i];
}

// ---------- WMMA fp32 matvec core: y = epilogue(W @ x) ----------
// W row-major MxK, K % 64 == 0, 256 threads = 8 waves, 128 rows per block.
// MODE: 0 plain, 1 sigmoid, 2 relu^2, 3 res+acc, 4 res+gate*acc
template <int MODE>
__device__ __forceinline__ void matvec_core(const float* __restrict__ W,
                                            const float* __restrict__ x,
                                            float* __restrict__ y,
                                            int M, int K, int bx,
                                            const float* __restrict__ res,
                                            const float* __restrict__ gate,
                                            float* sxv)
{
    stage_x_async(x, sxv, K);

    const int tid  = threadIdx.x;
    const int wave = tid >> 5;
    const int lane = tid & 31;
    const int m    = lane & 15;
    const int kh   = (lane >> 4) << 1;          // 0 (lanes 0-15) or 2 (16-31)
    const int rb   = bx * 128 + wave * 16;

    int rowc = rb + m;                          // clamp: EXEC stays all-1s
    if (rowc > M - 1) rowc = M - 1;
    const float* Ap = W + (size_t)rowc * K + kh;

    v8f acc = {0.f, 0.f, 0.f, 0.f, 0.f, 0.f, 0.f, 0.f};

    for (int k0 = 0; k0 < K; k0 += 64) {
        __builtin_prefetch(Ap + k0 + 256, 0, 1);
        #pragma unroll
        for (int u = 0; u < 16; ++u) {
            const int k = k0 + 4 * u;
            v2f a = *(const v2f*)(Ap + k);      // W[row][k+kh..k+kh+1]
            v2f bb;
            bb.x = sxv[k + kh];
            bb.y = sxv[k + kh + 1];
            acc = __builtin_amdgcn_wmma_f32_16x16x4_f32(
                false, a, false, bb, (short)0, acc, false, false);
        }
    }

    // D: VGPR r -> lanes 0-15: M=r, lanes 16-31: M=8+r; all columns equal.
    if ((lane & 15) == 0) {
        const int base = rb + ((lane >> 4) << 3);
        #pragma unroll
        for (int r = 0; r < 8; ++r) {
            const int row = base + r;
            if (row < M) {
                float v = acc[r];
                if (MODE == 1)      v = 1.0f / (1.0f + __expf(-v));
                else if (MODE == 2) { v = v > 0.0f ? v : 0.0f; v *= v; }
                else if (MODE == 3) v = res[row] + v;
                else if (MODE == 4) v = res[row] + gate[row] * v;
                y[row] = v;
            }
        }
    }
}

// =================== fused per-layer kernels ===================

// A: k|v|rpre = {kw,vw,rw} @ {xk,xv,xr}; last block runs the WKV recurrence.
__global__ __launch_bounds__(256)
void k_att_kvr(const float* __restrict__ kw, const float* __restrict__ vw,
               const float* __restrict__ rw, const float* __restrict__ xin,
               float* __restrict__ kvp,
               const float* __restrict__ tf, const float* __restrict__ td,
               const float* __restrict__ st, float* __restrict__ nst,
               float* __restrict__ rab, unsigned* cnt)
{
    __shared__ float sxv[4096];
    const int sel = blockIdx.y;
    const float* W = (sel == 0) ? kw : ((sel == 1) ? vw : rw);
    matvec_core<0>(W, xin + sel * Eh, kvp + sel * Eh, Eh, Eh, blockIdx.x,
                   nullptr, nullptr, sxv);
    if (!grid_sink(cnt, gridDim.x * gridDim.y)) return;

    for (int i = threadIdx.x; i < Eh; i += 256) {
        const float kk = kvp[i], vv = kvp[Eh + i];
        const float r  = 1.0f / (1.0f + __expf(-kvp[2 * Eh + i]));
        const float p0 = st[4 * Eh + i], A = st[2 * Eh + i], B = st[3 * Eh + i];
        const float ww = tf[i] + kk;
        const float p  = fmaxf(p0, ww);
        const float e1 = __expf(p0 - p), e2 = __expf(ww - p);
        rab[i] = r * (e1 * A + e2 * vv) / (e1 * B + e2);
        const float ww2 = p0 + td[i];
        const float p2  = fmaxf(ww2, kk);
        const float f1  = __expf(ww2 - p2), f2 = __expf(kk - p2);
        nst[2 * Eh + i] = f1 * A + f2 * vv;
        nst[3 * Eh + i] = f1 * B + f2;
        nst[4 * Eh + i] = p2;
    }
}

// B: sx = x + ow@rab; last block runs LN2 + channel-mix token shift.
__global__ __launch_bounds__(256)
void k_att_out(const float* __restrict__ ow, const float* __restrict__ rab,
               const float* __restrict__ x, float* __restrict__ x2,
               const float* __restrict__ l2w, const float* __restrict__ l2b,
               const float* __restrict__ ftmk, const float* __restrict__ ftmr,
               const float* __restrict__ st, float* __restrict__ nst,
               float* __restrict__ fxk, unsigned* cnt)
{
    __shared__ float sxv[4096];
    matvec_core<3>(ow, rab, x2, Eh, Eh, blockIdx.x, x, nullptr, sxv);
    if (!grid_sink(cnt, gridDim.x)) return;

    float xn[4];
    ln1024_t256(x2, l2w, l2b, xn);
    const int t = threadIdx.x;
    #pragma unroll
    for (int j = 0; j < 4; ++j) {
        const int i = t + 256 * j;
        const float fx = st[i];
        const float tk = ftmk[i], tr = ftmr[i];
        fxk[i]      = xn[j] * tk + fx * (1.0f - tk);
        fxk[Eh + i] = xn[j] * tr + fx * (1.0f - tr);
        nst[i]      = xn[j];                  // new ffn_x state
    }
}

// C: fk = relu(fkw@fxk)^2 (32 blocks) and fr = sigmoid(frw@fxr) (8 blocks).
__global__ __launch_bounds__(256)
void k_ffn_kr(const float* __restrict__ fkw, const float* __restrict__ frw,
              const float* __restrict__ fxk, float* __restrict__ fk,
              float* __restrict__ fr)
{
    __shared__ float sxv[4096];
    const int bx = blockIdx.x;
    if (bx < Hh / 128)
        matvec_core<2>(fkw, fxk, fk, Hh, Eh, bx, nullptr, nullptr, sxv);
    else
        matvec_core<1>(frw, fxk + Eh, fr, Eh, Eh, bx - Hh / 128,
                       nullptr, nullptr, sxv);
}

// D: x_next = sx + fr*(fvw@fk); last block preps the NEXT layer's LN1+mixes,
// or (last layer) the final LayerNorm.
__global__ __launch_bounds__(256)
void k_ffn_out(const float* __restrict__ fvw, const float* __restrict__ fk,
               const float* __restrict__ x2, const float* __restrict__ fr,
               float* __restrict__ x,
               const float* __restrict__ l1w, const float* __restrict__ l1b,
               const float* __restrict__ tmk, const float* __restrict__ tmv,
               const float* __restrict__ tmr,
               const float* __restrict__ st_att, float* __restrict__ nst_att,
               float* __restrict__ xk,
               const float* __restrict__ lnfw, const float* __restrict__ lnfb,
               float* __restrict__ xf, int is_last_layer, unsigned* cnt)
{
    __shared__ float sxv[4096];
    matvec_core<4>(fvw, fk, x, Eh, Hh, blockIdx.x, x2, fr, sxv);
    if (!grid_sink(cnt, gridDim.x)) return;

    const int t = threadIdx.x;
    float xn[4];
    if (!is_last_layer) {
        ln1024_t256(x, l1w, l1b, xn);
        #pragma unroll
        for (int j = 0; j < 4; ++j) {
            const int i = t + 256 * j;
            const float ax = st_att[i];
            const float tk = tmk[i], tv = tmv[i], tr = tmr[i];
            xk[i]          = xn[j] * tk + ax * (1.0f - tk);
            xk[Eh + i]     = xn[j] * tv + ax * (1.0f - tv);
            xk[2 * Eh + i] = xn[j] * tr + ax * (1.0f - tr);
            nst_att[i]     = xn[j];           // new att_x state
        }
    } else {
        ln1024_t256(x, lnfw, lnfb, xn);
        #pragma unroll
        for (int j = 0; j < 4; ++j) xf[t + 256 * j] = xn[j];
    }
}

// embedding + ln0 + layer-0 LN1/mixes (one 1024-thread block)
__global__ __launch_bounds__(1024)
void k_embed(const float* __restrict__ emb, const int* __restrict__ token,
             const float* __restrict__ ln0w, const float* __restrict__ ln0b,
             const float* __restrict__ l1w, const float* __restrict__ l1b,
             const float* __restrict__ tmk, const float* __restrict__ tmv,
             const float* __restrict__ tmr,
             const float* __restrict__ st_att, float* __restrict__ nst_att,
             float* __restrict__ x, float* __restrict__ xk)
{
    const int i = threadIdx.x;
    const float xi = emb[(size_t)token[0] * Eh + i];
    const float x0 = block_ln_1024(xi, ln0w, ln0b, i);
    x[i] = x0;
    const float xn = block_ln_1024(x0, l1w, l1b, i);
    const float ax = st_att[i];
    const float tk = tmk[i], tv = tmv[i], tr = tmr[i];
    xk[i]          = xn * tk + ax * (1.0f - tk);
    xk[Eh + i]     = xn * tv + ax * (1.0f - tv);
    xk[2 * Eh + i] = xn * tr + ax * (1.0f - tr);
    nst_att[i]     = xn;
}

// standalone matvec (head)
template <int MODE>
__global__ __launch_bounds__(256)
void matvec_wmma(const float* __restrict__ W, const float* __restrict__ x,
                 float* __restrict__ y, int M, int K,
                 const float* __restrict__ res, const float* __restrict__ gate)
{
    __shared__ float sxv[4096];
    matvec_core<MODE>(W, x, y, M, K, blockIdx.x, res, gate, sxv);
}

__global__ void k_init(unsigned* cnt) { cnt[threadIdx.x] = 0u; }

// =================== host orchestration (graph-capture safe) ===================
extern "C" void kernel_launch(void* const* d_in, const int* in_sizes, int n_in,
                              void* d_out, int out_size, void* d_ws, size_t ws_size,
                              hipStream_t stream)
{
    const float* emb      = (const float*)d_in[0];
    const float* ln0_w    = (const float*)d_in[1];
    const float* ln0_b    = (const float*)d_in[2];
    const float* ln1_w    = (const float*)d_in[3];
    const float* ln1_b    = (const float*)d_in[4];
    const float* att_tmk  = (const float*)d_in[5];
    const float* att_tmv  = (const float*)d_in[6];
    const float* att_tmr  = (const float*)d_in[7];
    const float* att_tf   = (const float*)d_in[8];
    const float* att_td   = (const float*)d_in[9];
    const float* att_kw   = (const float*)d_in[10];
    const float* att_vw   = (const float*)d_in[11];
    const float* att_rw   = (const float*)d_in[12];
    const float* att_ow   = (const float*)d_in[13];
    const float* ln2_w    = (const float*)d_in[14];
    const float* ln2_b    = (const float*)d_in[15];
    const float* ffn_tmk  = (const float*)d_in[16];
    const float* ffn_tmr  = (const float*)d_in[17];
    const float* ffn_kw   = (const float*)d_in[18];
    const float* ffn_vw   = (const float*)d_in[19];
    const float* ffn_rw   = (const float*)d_in[20];
    const float* ln_out_w = (const float*)d_in[21];
    const float* ln_out_b = (const float*)d_in[22];
    const float* head_w   = (const float*)d_in[23];
    const float* state    = (const float*)d_in[24];
    const int*   token    = (const int*)d_in[25];

    float* logits = (float*)d_out;                 // [V]
    float* nstate = logits + Vh;                   // [5*L*E]

    float* ws  = (float*)d_ws;
    float* x   = ws;                               // E
    float* x2  = ws + Eh;                          // E
    float* xk  = ws + 2 * Eh;                      // 3E  xk,xv,xr
    float* kvp = ws + 5 * Eh;                      // 3E  k,v,rpre
    float* rab = ws + 8 * Eh;                      // E
    float* fxk = ws + 9 * Eh;                      // 2E  fxk,fxr
    float* fr  = ws + 11 * Eh;                     // E
    float* xf  = ws + 12 * Eh;                     // E
    float* fk  = ws + 13 * Eh;                     // H
    unsigned* cnt = (unsigned*)(ws + 13 * Eh + Hh);// fan-in counters

    const size_t EE = (size_t)Eh * Eh;
    const size_t HE = (size_t)Hh * Eh;

    k_init<<<1, 8, 0, stream>>>(cnt);

    k_embed<<<1, 1024, 0, stream>>>(emb, token, ln0_w, ln0_b,
                                    ln1_w, ln1_b, att_tmk, att_tmv, att_tmr,
                                    state + Eh, nstate + Eh, x, xk);

    for (int l = 0; l < Lh; ++l) {
        const float* st  = state  + (size_t)l * 5 * Eh;  // [ffn_x,att_x,aa,bb,pp]
        float*       nst = nstate + (size_t)l * 5 * Eh;
        const int    ln  = (l < Lh - 1) ? (l + 1) : 0;   // next-layer params (dummy on last)

        k_att_kvr<<<dim3(Eh / 128, 3), 256, 0, stream>>>(
            att_kw + l * EE, att_vw + l * EE, att_rw + l * EE, xk, kvp,
            att_tf + l * Eh, att_td + l * Eh, st, nst, rab, cnt + 0);

        k_att_out<<<Eh / 128, 256, 0, stream>>>(
            att_ow + l * EE, rab, x, x2, ln2_w + l * Eh, ln2_b + l * Eh,
            ffn_tmk + l * Eh, ffn_tmr + l * Eh, st, nst, fxk, cnt + 1);

        k_ffn_kr<<<Hh / 128 + Eh / 128, 256, 0, stream>>>(
            ffn_kw + l * HE, ffn_rw + l * EE, fxk, fk, fr);

        k_ffn_out<<<Eh / 128, 256, 0, stream>>>(
            ffn_vw + l * HE, fk, x2, fr, x,
            ln1_w + ln * Eh, ln1_b + ln * Eh,
            att_tmk + ln * Eh, att_tmv + ln * Eh, att_tmr + ln * Eh,
            state + (size_t)ln * 5 * Eh + Eh, nstate + (size_t)ln * 5 * Eh + Eh,
            xk, ln_out_w, ln_out_b, xf, (l == Lh - 1) ? 1 : 0, cnt + 2);
    }

    // logits = head_w @ ln(x)
    matvec_wmma<0><<<(Vh + 127) / 128, 256, 0, stream>>>(head_w, xf, logits,
                                                         Vh, Eh, nullptr, nullptr);
}